// NetworkAnt_30958124270377
// MI455X (gfx1250) — compile-verified
//
#include <hip/hip_runtime.h>
#include <hip/hip_bf16.h>
#include <math.h>

#define N_NODES 4096
#define F_IN    64
#define H_DIM   128
#define NHEAD   4
#define HDHD    32
#define C_CLS   7
#define CAP     256   // max stored neighbors per row (E[deg]≈16.4, binom tail << CAP)

typedef float v2f __attribute__((ext_vector_type(2)));
typedef float v8f __attribute__((ext_vector_type(8)));

// ---------------------------------------------------------------------------
// fp32 WMMA GEMM: D = act(A(MxK) @ B(KxN) + bias), one wave = one 16x16 tile.
// K compile-time (full unroll); GUARD=false when N % 16 == 0 (no masking at all).
// Ragged-N path uses clamped index + multiply-mask => no EXEC divergence in loop.
// ---------------------------------------------------------------------------
template<int K, bool GUARD>
__global__ void gemm_f32_wmma(const float* __restrict__ A, int lda,
                              const float* __restrict__ B, int ldb,
                              const float* __restrict__ bias,
                              float* __restrict__ D, int ldc,
                              int M, int Ncols, int doRelu)
{
    int wid  = (blockIdx.x * blockDim.x + threadIdx.x) >> 5;
    int lane = threadIdx.x & 31;
    int mtiles = M >> 4;
    int ntiles = (Ncols + 15) >> 4;
    if (wid >= mtiles * ntiles) return;                 // whole-wave uniform exit
    int tm = wid % mtiles, tn = wid / mtiles;
    int r0 = tm << 4, c0 = tn << 4;
    int half = lane >> 4, m = lane & 15;                // A: lanes0-15 K={0,1}, 16-31 K={2,3}
    int col = c0 + m;

    bool  cok  = true;
    float mask = 1.0f;
    int   colc = col;
    if (GUARD) {
        cok  = (col < Ncols);
        mask = cok ? 1.0f : 0.0f;
        colc = cok ? col : (Ncols - 1);                 // always in-bounds load address
    }

    const float* Arow = A + (r0 + m) * lda + 2 * half;
    const float* Bcol = B + 2 * half * ldb + colc;

    v8f acc = {};
#pragma unroll
    for (int kb = 0; kb < K; kb += 4) {
        v2f a, b;
        a.x = Arow[kb];
        a.y = Arow[kb + 1];
        float bx = Bcol[kb * ldb];
        float by = Bcol[(kb + 1) * ldb];
        if (GUARD) { bx *= mask; by *= mask; }
        b.x = bx; b.y = by;
        acc = __builtin_amdgcn_wmma_f32_16x16x4_f32(false, a, false, b,
                                                    (short)0, acc, false, false);
    }
    if (cok) {
        float bv = bias ? bias[col] : 0.0f;
#pragma unroll
        for (int v = 0; v < 8; ++v) {                   // D row = r0 + 8*half + v, col per lane
            float val = acc[v] + bv;
            if (doRelu) val = fmaxf(val, 0.0f);
            D[(r0 + 8 * half + v) * ldc + col] = val;
        }
    }
}

// ---------------------------------------------------------------------------
// hh = h @ attn_W[l] (+Wb): B(k, n=head*32+d) = Wl[head][k][d]  (M=4096,K=128,N=128)
// ---------------------------------------------------------------------------
__global__ void gemm_hh_wmma(const float* __restrict__ h,
                             const float* __restrict__ Wl,   // NHEAD x H x HD
                             const float* __restrict__ Wbl,  // NHEAD x HD
                             float* __restrict__ hh)
{
    int wid  = (blockIdx.x * blockDim.x + threadIdx.x) >> 5;
    int lane = threadIdx.x & 31;
    const int mtiles = N_NODES / 16;
    if (wid >= mtiles * (H_DIM / 16)) return;
    int tm = wid % mtiles, tn = wid / mtiles;
    int r0 = tm << 4, c0 = tn << 4;
    int half = lane >> 4, m = lane & 15;
    int col = c0 + m;
    int head = col >> 5, d = col & 31;

    const float* Arow = h + (r0 + m) * H_DIM + 2 * half;
    const float* Bcol = Wl + head * (H_DIM * HDHD) + 2 * half * HDHD + d;

    v8f acc = {};
#pragma unroll
    for (int kb = 0; kb < H_DIM; kb += 4) {
        v2f a, b;
        a.x = Arow[kb];
        a.y = Arow[kb + 1];
        b.x = Bcol[kb * HDHD];
        b.y = Bcol[(kb + 1) * HDHD];
        acc = __builtin_amdgcn_wmma_f32_16x16x4_f32(false, a, false, b,
                                                    (short)0, acc, false, false);
    }
    float bv = Wbl[head * HDHD + d];
#pragma unroll
    for (int v = 0; v < 8; ++v)
        hh[(r0 + 8 * half + v) * H_DIM + col] = acc[v] + bv;
}

// ---------------------------------------------------------------------------
// One pass over adj: per-row degree, dinv = rsqrt(rowsum+1), CSR column list.
// ---------------------------------------------------------------------------
__global__ void build_graph(const float* __restrict__ adj,
                            float* __restrict__ dinv,
                            int* __restrict__ cnt_cap,
                            int* __restrict__ cnt_true,
                            int* __restrict__ cols)
{
    __shared__ int   s_cnt;
    __shared__ float s_sum[256];
    int row = blockIdx.x;
    int t   = threadIdx.x;
    if (t == 0) s_cnt = 0;
    __syncthreads();
    const float* arow = adj + (size_t)row * N_NODES;
    float sum = 0.f;
    for (int j = t; j < N_NODES; j += 256) {
        float v = arow[j];
        sum += v;
        if (v > 0.f) {
            int k = atomicAdd(&s_cnt, 1);
            if (k < CAP) cols[(size_t)row * CAP + k] = j;
        }
    }
    s_sum[t] = sum;
    __syncthreads();
    for (int off = 128; off > 0; off >>= 1) {
        if (t < off) s_sum[t] += s_sum[t + off];
        __syncthreads();
    }
    if (t == 0) {
        cnt_true[row] = s_cnt;
        cnt_cap[row]  = s_cnt < CAP ? s_cnt : CAP;
        dinv[row]     = rsqrtf(s_sum[0] + 1.0f);
    }
}

// ---------------------------------------------------------------------------
// Sparse GCN aggregation: agg_i = dinv_i*(dinv_i*h_i + sum_{j in N(i)} dinv_j*h_j)
// ---------------------------------------------------------------------------
__global__ void gcn_agg(const float* __restrict__ h,
                        const float* __restrict__ dinv,
                        const int* __restrict__ cnt_cap,
                        const int* __restrict__ cols,
                        float* __restrict__ agg)
{
    __shared__ int   s_cols[CAP];
    __shared__ float s_w[CAP];
    int i = blockIdx.x, c = threadIdx.x;   // 128 threads
    int cnt = cnt_cap[i];
    for (int k = c; k < cnt; k += 128) {
        int j = cols[(size_t)i * CAP + k];
        s_cols[k] = j;
        s_w[k]    = dinv[j];
    }
    __syncthreads();
    float di  = dinv[i];
    float acc = di * h[i * H_DIM + c];     // identity self-loop term
    for (int k = 0; k < cnt; ++k)
        acc += s_w[k] * h[s_cols[k] * H_DIM + c];
    agg[i * H_DIM + c] = di * acc;
}

// ---------------------------------------------------------------------------
// Per-node per-head attention logits (hh already includes Wb).
// ---------------------------------------------------------------------------
__global__ void attn_scores(const float* __restrict__ hh,
                            const float* __restrict__ a_l,   // NHEAD x 2*HD
                            float* __restrict__ s_src,
                            float* __restrict__ s_dst)
{
    int idx = blockIdx.x * blockDim.x + threadIdx.x;
    if (idx >= N_NODES * NHEAD) return;
    int head = idx / N_NODES;
    int n    = idx % N_NODES;
    const float* hv = hh + n * H_DIM + head * HDHD;
    const float* av = a_l + head * 2 * HDHD;
    float ss = 0.f, sd = 0.f;
#pragma unroll
    for (int d = 0; d < HDHD; ++d) {
        float x = hv[d];
        ss += x * av[d];
        sd += x * av[HDHD + d];
    }
    s_src[head * N_NODES + n] = ss;
    s_dst[head * N_NODES + n] = sd;
}

__global__ void zero_vec(float* p, int n)
{
    int i = blockIdx.x * blockDim.x + threadIdx.x;
    if (i < n) p[i] = 0.f;
}

// column sums of a (rows x 128) matrix into out[128]
__global__ void colsum128(const float* __restrict__ Mx, int rows, float* __restrict__ out)
{
    int c = threadIdx.x;                   // 128 threads
    int chunk = rows / gridDim.x;
    int r0 = blockIdx.x * chunk;
    float acc = 0.f;
    for (int r = r0; r < r0 + chunk; ++r) acc += Mx[r * H_DIM + c];
    atomicAdd(&out[c], acc);
}

// ---------------------------------------------------------------------------
// Sparse attention (masked-with-0 softmax collapsed) fused with GCN combine:
//   Z_i   = (N - deg_i) + sum_edges exp(s)
//   att_i = (sum_edges (e-1)*hh_j + colsum(hh)) / Z_i
//   h_i   = relu(g_i + att_i)
// ---------------------------------------------------------------------------
__global__ void attn_combine(const float* __restrict__ hh,
                             const float* __restrict__ g,
                             const float* __restrict__ s_src,
                             const float* __restrict__ s_dst,
                             const float* __restrict__ ab,     // NHEAD
                             const float* __restrict__ csum,   // 128
                             const int* __restrict__ cnt_cap,
                             const int* __restrict__ cnt_true,
                             const int* __restrict__ cols,
                             float* __restrict__ h_out)
{
    __shared__ int s_cols[CAP];
    int i = blockIdx.x, c = threadIdx.x;   // 128 threads
    int head = c >> 5;
    int cnt = cnt_cap[i];
    for (int k = c; k < cnt; k += 128) s_cols[k] = cols[(size_t)i * CAP + k];
    __syncthreads();
    float si  = s_src[head * N_NODES + i] + ab[head];
    float acc = 0.f, esum = 0.f;
    for (int k = 0; k < cnt; ++k) {
        int j   = s_cols[k];
        float e = expf(si + s_dst[head * N_NODES + j]) - 1.0f;
        esum += e;
        acc  += e * hh[j * H_DIM + c];
    }
    float Z = (float)(N_NODES - cnt_true[i]) + (float)cnt + esum;
    float att = (acc + csum[c]) / Z;
    h_out[i * H_DIM + c] = fmaxf(g[i * H_DIM + c] + att, 0.f);
}

// ---------------------------------------------------------------------------
// contagion head from column-sums of final h (64 threads, one block)
// ---------------------------------------------------------------------------
__global__ void contagion_kernel(const float* __restrict__ csum_h,  // 128
                                 const float* __restrict__ W1, const float* __restrict__ b1,
                                 const float* __restrict__ W2, const float* __restrict__ b2,
                                 float* __restrict__ out)
{
    __shared__ float s_mean[H_DIM];
    __shared__ float s_hid[64];
    int t = threadIdx.x;                   // 64 threads
    s_mean[t]      = csum_h[t]      * (1.0f / N_NODES);
    s_mean[t + 64] = csum_h[t + 64] * (1.0f / N_NODES);
    __syncthreads();
    float acc = b1[t];
#pragma unroll
    for (int f = 0; f < H_DIM; ++f) acc += s_mean[f] * W1[f * 64 + t];
    s_hid[t] = fmaxf(acc, 0.f) * W2[t];
    __syncthreads();
    for (int off = 32; off > 0; off >>= 1) {
        if (t < off) s_hid[t] += s_hid[t + off];
        __syncthreads();
    }
    if (t == 0) out[0] = s_hid[0] + b2[0];
}

// ---------------------------------------------------------------------------
extern "C" void kernel_launch(void* const* d_in, const int* in_sizes, int n_in,
                              void* d_out, int out_size, void* d_ws, size_t ws_size,
                              hipStream_t stream)
{
    const float* x       = (const float*)d_in[0];
    const float* adj     = (const float*)d_in[1];
    const float* enc_W   = (const float*)d_in[2];
    const float* enc_b   = (const float*)d_in[3];
    const float* gcn_W   = (const float*)d_in[4];
    const float* gcn_b   = (const float*)d_in[5];
    const float* attn_W  = (const float*)d_in[6];
    const float* attn_Wb = (const float*)d_in[7];
    const float* attn_a  = (const float*)d_in[8];
    const float* attn_ab = (const float*)d_in[9];
    const float* cls_W1  = (const float*)d_in[10];
    const float* cls_b1  = (const float*)d_in[11];
    const float* cls_W2  = (const float*)d_in[12];
    const float* cls_b2  = (const float*)d_in[13];
    const float* con_W1  = (const float*)d_in[14];
    const float* con_b1  = (const float*)d_in[15];
    const float* con_W2  = (const float*)d_in[16];
    const float* con_b2  = (const float*)d_in[17];
    float* out = (float*)d_out;

    // workspace carve-up
    char* ws = (char*)d_ws;
    size_t off = 0;
    auto carve = [&](size_t bytes) -> void* {
        void* p = ws + off;
        off += (bytes + 255) & ~(size_t)255;
        return p;
    };
    float* dinv     = (float*)carve((size_t)N_NODES * 4);
    int*   cnt_cap  = (int*)  carve((size_t)N_NODES * 4);
    int*   cnt_true = (int*)  carve((size_t)N_NODES * 4);
    int*   cols     = (int*)  carve((size_t)N_NODES * CAP * 4);
    float* h        = (float*)carve((size_t)N_NODES * H_DIM * 4);
    float* agg      = (float*)carve((size_t)N_NODES * H_DIM * 4);
    float* g        = (float*)carve((size_t)N_NODES * H_DIM * 4);
    float* hh       = (float*)carve((size_t)N_NODES * H_DIM * 4);
    float* ssrc     = (float*)carve((size_t)NHEAD * N_NODES * 4);
    float* sdst     = (float*)carve((size_t)NHEAD * N_NODES * 4);
    float* csum     = (float*)carve((size_t)H_DIM * 4);
    float* t1       = (float*)carve((size_t)N_NODES * 64 * 4);
    (void)ws_size; (void)in_sizes; (void)n_in; (void)out_size;

    auto blocks_for = [](int M, int Nc) {
        int tiles = (M / 16) * ((Nc + 15) / 16);
        return (tiles * 32 + 127) / 128;
    };

    // graph preprocessing: one streaming pass over the 64MB adjacency
    build_graph<<<N_NODES, 256, 0, stream>>>(adj, dinv, cnt_cap, cnt_true, cols);

    // encoder: h = relu(x @ enc_W + enc_b)   (M=4096, K=64, N=128)
    gemm_f32_wmma<F_IN, false><<<blocks_for(N_NODES, H_DIM), 128, 0, stream>>>(
        x, F_IN, enc_W, H_DIM, enc_b, h, H_DIM, N_NODES, H_DIM, 1);

    for (int l = 0; l < 3; ++l) {
        // GCN branch: sparse aggregate, then dense 128x128 projection + relu
        gcn_agg<<<N_NODES, 128, 0, stream>>>(h, dinv, cnt_cap, cols, agg);
        gemm_f32_wmma<H_DIM, false><<<blocks_for(N_NODES, H_DIM), 128, 0, stream>>>(
            agg, H_DIM, gcn_W + (size_t)l * H_DIM * H_DIM, H_DIM,
            gcn_b + (size_t)l * H_DIM, g, H_DIM, N_NODES, H_DIM, 1);

        // attention branch
        gemm_hh_wmma<<<blocks_for(N_NODES, H_DIM), 128, 0, stream>>>(
            h, attn_W + (size_t)l * NHEAD * H_DIM * HDHD,
            attn_Wb + (size_t)l * NHEAD * HDHD, hh);
        attn_scores<<<(N_NODES * NHEAD + 127) / 128, 128, 0, stream>>>(
            hh, attn_a + (size_t)l * NHEAD * 2 * HDHD, ssrc, sdst);
        zero_vec<<<1, 128, 0, stream>>>(csum, H_DIM);
        colsum128<<<32, 128, 0, stream>>>(hh, N_NODES, csum);
        attn_combine<<<N_NODES, 128, 0, stream>>>(
            hh, g, ssrc, sdst, attn_ab + (size_t)l * NHEAD, csum,
            cnt_cap, cnt_true, cols, h);
    }

    // node classifier -> out[0 : 4096*7)
    gemm_f32_wmma<H_DIM, false><<<blocks_for(N_NODES, 64), 128, 0, stream>>>(
        h, H_DIM, cls_W1, 64, cls_b1, t1, 64, N_NODES, 64, 1);
    gemm_f32_wmma<64, true><<<blocks_for(N_NODES, C_CLS), 128, 0, stream>>>(
        t1, 64, cls_W2, C_CLS, cls_b2, out, C_CLS, N_NODES, C_CLS, 0);

    // h -> out[4096*7 : 4096*7 + 4096*128)
    hipMemcpyAsync(out + (size_t)N_NODES * C_CLS, h,
                   (size_t)N_NODES * H_DIM * sizeof(float),
                   hipMemcpyDeviceToDevice, stream);

    // contagion head -> last element
    zero_vec<<<1, 128, 0, stream>>>(csum, H_DIM);
    colsum128<<<32, 128, 0, stream>>>(h, N_NODES, csum);
    contagion_kernel<<<1, 64, 0, stream>>>(csum, con_W1, con_b1, con_W2, con_b2,
                                           out + (size_t)N_NODES * C_CLS + (size_t)N_NODES * H_DIM);
}